// GraphAutoEncoder_48198122996338
// MI455X (gfx1250) — compile-verified
//
#include <hip/hip_runtime.h>
#include <hip/hip_bf16.h>

// ---------------------------------------------------------------------------
// GraphAutoEncoder forward, fused: one wave32 per graph (B=4096, N=12, H=64).
// All matmuls >= K=4 run on V_WMMA_F32_16X16X4_F32 (full fp32 precision).
// Activations staged in LDS; weights streamed from global (L2-resident).
// ---------------------------------------------------------------------------

typedef float v2f __attribute__((ext_vector_type(2)));
typedef float v8f __attribute__((ext_vector_type(8)));

#define LDX 68   // stride for 16x64 activation tiles (even -> aligned b64 pair loads)
#define LDE 20   // stride for 16x16 attention tile
#define SLOPE 0.2f
#define ALPHA 0.1f

__device__ __forceinline__ v8f wmma4(v2f a, v2f b, v8f c) {
  // D = A(16x4) * B(4x16) + C, fp32 everywhere.
  return __builtin_amdgcn_wmma_f32_16x16x4_f32(false, a, false, b, (short)0, c,
                                               false, false);
}

__device__ __forceinline__ void zero4(v8f a[4]) {
#pragma unroll
  for (int t = 0; t < 4; ++t)
#pragma unroll
    for (int r = 0; r < 8; ++r) a[t][r] = 0.f;
}

// C(16x64) += X_lds(16xK, stride LDX) @ W_global(Kx64 row-major)
__device__ __forceinline__ void mm_X_Wg(const float* X, const float* W, int K,
                                        int lo, int hi, v8f acc[4]) {
  for (int k = 0; k < K; k += 4) {
    const int ka = k + 2 * hi;
    v2f a; a[0] = X[lo * LDX + ka]; a[1] = X[lo * LDX + ka + 1];
    const float* wr = W + ka * 64 + lo;
#pragma unroll
    for (int t = 0; t < 4; ++t) {
      v2f b; b[0] = wr[16 * t]; b[1] = wr[64 + 16 * t];
      acc[t] = wmma4(a, b, acc[t]);
    }
  }
}

// C(16x64) += E_lds(16x16, stride LDE) @ XL_lds(16x64, stride LDX)
__device__ __forceinline__ void agg_mm(const float* E, const float* XL,
                                       int lo, int hi, v8f acc[4]) {
#pragma unroll
  for (int k = 0; k < 16; k += 4) {
    const int ka = k + 2 * hi;
    v2f a; a[0] = E[lo * LDE + ka]; a[1] = E[lo * LDE + ka + 1];
#pragma unroll
    for (int t = 0; t < 4; ++t) {
      v2f b; b[0] = XL[ka * LDX + 16 * t + lo];
      b[1] = XL[(ka + 1) * LDX + 16 * t + lo];
      acc[t] = wmma4(a, b, acc[t]);
    }
  }
}

// store C-layout accumulators into a 16x64 LDS tile
__device__ __forceinline__ void st16x64(float* Xd, int lo, int hi, const v8f acc[4]) {
#pragma unroll
  for (int t = 0; t < 4; ++t)
#pragma unroll
    for (int r = 0; r < 8; ++r)
      Xd[(r + 8 * hi) * LDX + 16 * t + lo] = acc[t][r];
}

__device__ __forceinline__ void bias_relu(v8f acc[4], const float* b, int lo) {
#pragma unroll
  for (int t = 0; t < 4; ++t) {
    const float bb = b[16 * t + lo];
#pragma unroll
    for (int r = 0; r < 8; ++r) acc[t][r] = fmaxf(acc[t][r] + bb, 0.f);
  }
}

// GATv2 attention: scores + masked softmax; result (a) left in E (zero-padded).
__device__ __forceinline__ void attn_softmax(int lane, const float* XL,
                                             const float* XR, const float* att,
                                             const unsigned* adj, float* E) {
  for (int p = lane; p < 144; p += 32) {
    const int i = p / 12, j = p - 12 * i;
    float e = 0.f;
    for (int h = 0; h < 64; ++h) {
      const float s = XL[j * LDX + h] + XR[i * LDX + h];
      e = fmaf(att[h], fmaxf(s, SLOPE * s), e);   // leaky_relu, slope<1
    }
    E[i * LDE + j] = e;
  }
  __syncthreads();
  if (lane < 16) {
    if (lane < 12) {
      const unsigned m = adj[lane];
      float mx = -3.0e38f;
      for (int j = 0; j < 12; ++j)
        if ((m >> j) & 1u) mx = fmaxf(mx, E[lane * LDE + j]);
      float av[16]; float s = 0.f;
#pragma unroll
      for (int j = 0; j < 16; ++j) {
        float a = 0.f;
        if (j < 12 && ((m >> j) & 1u)) a = __expf(E[lane * LDE + j] - mx);
        av[j] = a; s += a;
      }
      const float inv = 1.f / s;
#pragma unroll
      for (int j = 0; j < 16; ++j) E[lane * LDE + j] = av[j] * inv;
    } else {
      for (int j = 0; j < 16; ++j) E[lane * LDE + j] = 0.f;  // pad rows
    }
  }
  __syncthreads();
}

__global__ __launch_bounds__(32) void gae_fused_kernel(
    const float* __restrict__ batch,
    const float* __restrict__ enc_w1, const float* __restrict__ enc_b1,
    const float* __restrict__ enc_w2, const float* __restrict__ enc_b2,
    const float* __restrict__ enc_w3, const float* __restrict__ enc_b3,
    const float* __restrict__ g1_wl, const float* __restrict__ g1_wr,
    const float* __restrict__ g1_att, const float* __restrict__ g1_b,
    const float* __restrict__ g2_wl, const float* __restrict__ g2_wr,
    const float* __restrict__ g2_att, const float* __restrict__ g2_b,
    const float* __restrict__ g3_wl, const float* __restrict__ g3_wr,
    const float* __restrict__ g3_att, const float* __restrict__ g3_b,
    const float* __restrict__ g4_wl, const float* __restrict__ g4_wr,
    const float* __restrict__ g4_att, const float* __restrict__ g4_b,
    const float* __restrict__ lab_w, const float* __restrict__ lab_b,
    const float* __restrict__ val_w, const float* __restrict__ val_b,
    const float* __restrict__ skip_w, const float* __restrict__ skip_b,
    float* __restrict__ out, int B) {
  __shared__ float sX[16 * LDX];    // current activations (h1 -> h2 -> x1 -> x2)
  __shared__ float sXL[16 * LDX];   // xl / staging for x3,x4
  __shared__ float sXR[16 * LDX];   // xr
  __shared__ float sE[16 * LDE];    // attention scores / coefficients
  __shared__ float sLat[16 * 4];    // latent (12x3, zero-padded)
  __shared__ float sAtt[64];        // per-layer attention vector
  __shared__ float sWsk[4 * 64];    // skip_w zero-padded to 4x64
  __shared__ unsigned sAdj[16];     // adjacency bitmasks per row

  const int lane = threadIdx.x;
  const int lo = lane & 15, hi = lane >> 4;
  const int g = blockIdx.x;
  const float* bt = batch + (size_t)g * 12 * 5;

  // output sections (flat, in reference return order)
  float* out0 = out;                                 // batch[:,:,:4]
  float* out1 = out + (size_t)B * 48;                // batch[:,:,4:5]
  float* out2 = out1 + (size_t)B * 12;               // logits (B,12,4)
  float* out3 = out2 + (size_t)B * 48;               // values (B,12,1)
  float* out4 = out3 + (size_t)B * 12;               // latent (B,12,3)

  // ---- pass-through outputs + stage padded skip_w ----
  for (int idx = lane; idx < 60; idx += 32) {
    const int i = idx / 5, c = idx - 5 * i;
    const float v = bt[i * 5 + c];
    if (c < 4) out0[((size_t)g * 12 + i) * 4 + c] = v;
    else       out1[(size_t)g * 12 + i] = v;
  }
  for (int idx = lane; idx < 256; idx += 32)
    sWsk[idx] = (idx < 192) ? skip_w[idx] : 0.f;

  // ---- encoder layer 1 (K=5, scalar): h1 = relu(batch@w1+b1) -> sX ----
  for (int idx = lane; idx < 16 * 64; idx += 32) {
    const int i = idx >> 6, h = idx & 63;
    float s = 0.f;
    if (i < 12) {
      s = enc_b1[h];
      for (int c = 0; c < 5; ++c) s = fmaf(bt[i * 5 + c], enc_w1[c * 64 + h], s);
      s = fmaxf(s, 0.f);
    }
    sX[i * LDX + h] = s;   // pad rows 12..15 zeroed
  }
  __syncthreads();

  // ---- encoder layer 2 (WMMA): h2 = relu(h1@w2+b2) ----
  {
    v8f acc[4]; zero4(acc);
    mm_X_Wg(sX, enc_w2, 64, lo, hi, acc);
    bias_relu(acc, enc_b2, lo);
    __syncthreads();
    st16x64(sX, lo, hi, acc);   // sX <- h2
    __syncthreads();
  }

  // ---- encoder layer 3 (scalar, 64->3): latent -> sLat + out4 ----
  if (lane < 16) { sLat[lane*4+0]=0.f; sLat[lane*4+1]=0.f; sLat[lane*4+2]=0.f; sLat[lane*4+3]=0.f; }
  __syncthreads();
  for (int idx = lane; idx < 36; idx += 32) {
    const int i = idx / 3, c = idx - 3 * i;
    float s = enc_b3[c];
    for (int h = 0; h < 64; ++h) s = fmaf(sX[i * LDX + h], enc_w3[h * 3 + c], s);
    sLat[i * 4 + c] = s;
    out4[((size_t)g * 12 + i) * 3 + c] = s;
  }
  __syncthreads();

  // ---- Gabriel adjacency from latent[:, :2] (bitmask per row) ----
  if (lane < 12) {
    const float px = sLat[lane * 4], py = sLat[lane * 4 + 1];
    unsigned m = 0;
    for (int j = 0; j < 12; ++j) {
      if (j == lane) { m |= 1u << j; continue; }
      const float qx = sLat[j * 4], qy = sLat[j * 4 + 1];
      const float mx = (px + qx) * 0.5f, my = (py + qy) * 0.5f;
      const float dx = px - qx, dy = py - qy;
      const float r2 = (dx * dx + dy * dy) * 0.25f;
      bool ok = true;
      for (int k = 0; k < 12; ++k) {
        if (k == lane || k == j) continue;
        const float kx = sLat[k * 4] - mx, ky = sLat[k * 4 + 1] - my;
        ok = ok && (kx * kx + ky * ky > r2);
      }
      if (ok) m |= 1u << j;
    }
    sAdj[lane] = m;
  } else if (lane < 16) {
    sAdj[lane] = 0u;
  }
  __syncthreads();

  // ---- skip = ALPHA*(latent@skip_w + skip_b), kept in registers (WMMA K=4) ----
  v8f sk[4]; zero4(sk);
  {
    const int ka = 2 * hi;
    v2f a; a[0] = sLat[lo * 4 + ka]; a[1] = sLat[lo * 4 + ka + 1];
#pragma unroll
    for (int t = 0; t < 4; ++t) {
      v2f b; b[0] = sWsk[ka * 64 + 16 * t + lo];
      b[1] = sWsk[(ka + 1) * 64 + 16 * t + lo];
      sk[t] = wmma4(a, b, sk[t]);
    }
#pragma unroll
    for (int t = 0; t < 4; ++t) {
      const float bb = skip_b[16 * t + lo];
#pragma unroll
      for (int r = 0; r < 8; ++r) sk[t][r] = ALPHA * (sk[t][r] + bb);
    }
  }

  // ---- GAT layer 1 (cin=1): xl/xr are outer products of x0=latent[:,2] ----
  for (int idx = lane; idx < 16 * 64; idx += 32) {
    const int i = idx >> 6, h = idx & 63;
    const float x0 = (i < 12) ? sLat[i * 4 + 2] : 0.f;
    sXL[i * LDX + h] = x0 * g1_wl[h];
    sXR[i * LDX + h] = x0 * g1_wr[h];
  }
  for (int idx = lane; idx < 64; idx += 32) sAtt[idx] = g1_att[idx];
  __syncthreads();
  attn_softmax(lane, sXL, sXR, sAtt, sAdj, sE);
  {
    v8f acc[4]; zero4(acc);
    agg_mm(sE, sXL, lo, hi, acc);
    bias_relu(acc, g1_b, lo);
    __syncthreads();
    st16x64(sX, lo, hi, acc);   // sX <- x1
    __syncthreads();
  }

  // ---- GAT layer 2 ----
  {
    v8f axl[4], axr[4]; zero4(axl); zero4(axr);
    mm_X_Wg(sX, g2_wl, 64, lo, hi, axl);
    mm_X_Wg(sX, g2_wr, 64, lo, hi, axr);
    for (int idx = lane; idx < 64; idx += 32) sAtt[idx] = g2_att[idx];
    __syncthreads();
    st16x64(sXL, lo, hi, axl);
    st16x64(sXR, lo, hi, axr);
    __syncthreads();
    attn_softmax(lane, sXL, sXR, sAtt, sAdj, sE);
    v8f acc[4]; zero4(acc);
    agg_mm(sE, sXL, lo, hi, acc);
    bias_relu(acc, g2_b, lo);
    __syncthreads();
    st16x64(sX, lo, hi, acc);   // sX <- x2 (kept for g3 AND g4)
    __syncthreads();
  }

  // ---- GAT layer 3 -> x3 -> logits ----
  {
    v8f axl[4], axr[4]; zero4(axl); zero4(axr);
    mm_X_Wg(sX, g3_wl, 64, lo, hi, axl);
    mm_X_Wg(sX, g3_wr, 64, lo, hi, axr);
    for (int idx = lane; idx < 64; idx += 32) sAtt[idx] = g3_att[idx];
    __syncthreads();
    st16x64(sXL, lo, hi, axl);
    st16x64(sXR, lo, hi, axr);
    __syncthreads();
    attn_softmax(lane, sXL, sXR, sAtt, sAdj, sE);
    v8f acc[4]; zero4(acc);
    agg_mm(sE, sXL, lo, hi, acc);
#pragma unroll
    for (int t = 0; t < 4; ++t) {
      const float bb = g3_b[16 * t + lo];
#pragma unroll
      for (int r = 0; r < 8; ++r)
        acc[t][r] = fmaxf(acc[t][r] + bb + sk[t][r], 0.f);
    }
    __syncthreads();
    st16x64(sXL, lo, hi, acc);  // stage x3 (xl3 no longer needed)
    __syncthreads();
    // logits = x3 @ lab_w + lab_b (12x4, scalar)
    for (int idx = lane; idx < 48; idx += 32) {
      const int i = idx >> 2, c = idx & 3;
      float s = lab_b[c];
      for (int h = 0; h < 64; ++h) s = fmaf(sXL[i * LDX + h], lab_w[h * 4 + c], s);
      out2[((size_t)g * 12 + i) * 4 + c] = s;
    }
    __syncthreads();
  }

  // ---- GAT layer 4 -> x4 -> values ----
  {
    v8f axl[4], axr[4]; zero4(axl); zero4(axr);
    mm_X_Wg(sX, g4_wl, 64, lo, hi, axl);   // sX still holds x2
    mm_X_Wg(sX, g4_wr, 64, lo, hi, axr);
    for (int idx = lane; idx < 64; idx += 32) sAtt[idx] = g4_att[idx];
    __syncthreads();
    st16x64(sXL, lo, hi, axl);
    st16x64(sXR, lo, hi, axr);
    __syncthreads();
    attn_softmax(lane, sXL, sXR, sAtt, sAdj, sE);
    v8f acc[4]; zero4(acc);
    agg_mm(sE, sXL, lo, hi, acc);
#pragma unroll
    for (int t = 0; t < 4; ++t) {
      const float bb = g4_b[16 * t + lo];
#pragma unroll
      for (int r = 0; r < 8; ++r)
        acc[t][r] = fmaxf(acc[t][r] + bb + sk[t][r], 0.f);
    }
    __syncthreads();
    st16x64(sXL, lo, hi, acc);  // stage x4
    __syncthreads();
    // values = x4 @ val_w + val_b (12x1, scalar)
    for (int idx = lane; idx < 12; idx += 32) {
      float s = val_b[0];
      for (int h = 0; h < 64; ++h) s = fmaf(sXL[idx * LDX + h], val_w[h], s);
      out3[(size_t)g * 12 + idx] = s;
    }
  }
}

extern "C" void kernel_launch(void* const* d_in, const int* in_sizes, int n_in,
                              void* d_out, int out_size, void* d_ws, size_t ws_size,
                              hipStream_t stream) {
  (void)n_in; (void)out_size; (void)d_ws; (void)ws_size;
  const int B = in_sizes[0] / 60;   // batch shape (B, 12, 5)
  gae_fused_kernel<<<dim3(B), dim3(32), 0, stream>>>(
      (const float*)d_in[0],
      (const float*)d_in[1],  (const float*)d_in[2],
      (const float*)d_in[3],  (const float*)d_in[4],
      (const float*)d_in[5],  (const float*)d_in[6],
      (const float*)d_in[7],  (const float*)d_in[8],
      (const float*)d_in[9],  (const float*)d_in[10],
      (const float*)d_in[11], (const float*)d_in[12],
      (const float*)d_in[13], (const float*)d_in[14],
      (const float*)d_in[15], (const float*)d_in[16],
      (const float*)d_in[17], (const float*)d_in[18],
      (const float*)d_in[19], (const float*)d_in[20],
      (const float*)d_in[21], (const float*)d_in[22],
      (const float*)d_in[23], (const float*)d_in[24],
      (const float*)d_in[25], (const float*)d_in[26],
      (const float*)d_in[27], (const float*)d_in[28],
      (float*)d_out, B);
}